// PredictHandler_22668837388477
// MI455X (gfx1250) — compile-verified
//
#include <hip/hip_runtime.h>
#include <stdint.h>

#define THR_CONF 0.5f
#define THR_NMS  0.3f
#define KTOP     4096
#define CAND     8192
#define NBINS    (1u << 20)
#define IMG_OFF  1000.0f

typedef float        v2f __attribute__((ext_vector_type(2)));
typedef float        v8f __attribute__((ext_vector_type(8)));
typedef unsigned int v4u __attribute__((ext_vector_type(4)));
typedef int          v8i __attribute__((ext_vector_type(8)));
typedef int          v4i __attribute__((ext_vector_type(4)));

// ---------------- workspace byte offsets ----------------
// scores:   [0, 16 MiB)                        float[N]
// hist:     +16,777,216  (4 MiB)               uint[NBINS]
// meta:     +20,971,520  (64 B)                uint[16]  {T, cAbove, cntHi, cntEq}
// cand:     +20,971,584  (64 KiB)              u64[CAND] key = bits<<32 | ~idx
// top:      +21,037,120  (32 KiB)              uint2[KTOP] {scoreBits, flatIdx}
// nmsboxes: +21,069,888  (64 KiB)              float[KTOP*4] (offset ltrb)
// mask:     +21,135,424  (2 MiB)               u64[KTOP*64] suppression bits
#define OFF_SCORES 0u
#define OFF_HIST   16777216u
#define OFF_META   20971520u
#define OFF_CAND   20971584u
#define OFF_TOP    21037120u
#define OFF_NMSB   21069888u
#define OFF_MASK   21135424u

// ---------------- K0: zero hist+meta+cand (contiguous) ----------------
__global__ void k_init(uint32_t* __restrict__ w, int nwords) {
    int i = blockIdx.x * blockDim.x + threadIdx.x;
    if (i < nwords) w[i] = 0u;
}

// ---------------- K1: score = sigmoid(c1-c0), mask, histogram ----------------
__global__ void k_score_hist(const float* __restrict__ p_conf,
                             float* __restrict__ scores,
                             uint32_t* __restrict__ hist, int N) {
    int i = blockIdx.x * blockDim.x + threadIdx.x;
    if (i >= N) return;
    float c0 = p_conf[2 * i], c1 = p_conf[2 * i + 1];
    float s = 1.0f / (1.0f + expf(c0 - c1));   // softmax fg == sigmoid(c1-c0)
    if (s >= THR_CONF) {
        scores[i] = s;
        uint32_t b = __float_as_uint(s);            // in [0x3F000000, 0x3F800000]
        uint32_t bin = (b - 0x3F000000u) >> 3;      // 2^20 bins over [0.5, 1)
        if (bin >= NBINS) bin = NBINS - 1u;
        atomicAdd(&hist[bin], 1u);
    } else {
        scores[i] = 0.0f;
    }
}

// ---------------- K2: find threshold bin T containing the K-th score ----------------
__global__ void k_thresh(const uint32_t* __restrict__ hist, uint32_t* __restrict__ meta) {
    __shared__ uint32_t csum[1024];
    const uint32_t CH = NBINS / 1024u;
    int t = threadIdx.x;
    uint32_t s = 0;
    for (uint32_t b = (uint32_t)t * CH; b < (uint32_t)(t + 1) * CH; ++b) s += hist[b];
    csum[t] = s;
    __syncthreads();
    if (t == 0) {
        unsigned long long running = 0;
        int fc = -1;
        for (int c = 1023; c >= 0; --c) {
            if (running + csum[c] >= (unsigned long long)KTOP) { fc = c; break; }
            running += csum[c];
        }
        uint32_t T = 0; unsigned long long cAbove = 0;
        if (fc >= 0) {
            uint32_t base = (uint32_t)fc * CH;
            for (int b = (int)CH - 1; b >= 0; --b) {
                uint32_t h = hist[base + (uint32_t)b];
                if (running + h >= (unsigned long long)KTOP) {
                    T = base + (uint32_t)b; cAbove = running; break;
                }
                running += h;
            }
        } else {                       // fewer than K positives: take them all
            T = 0; cAbove = running - hist[0];
        }
        meta[0] = T;
        meta[1] = (uint32_t)cAbove;
    }
}

// ---------------- K3: compact candidates with bin >= T ----------------
__global__ void k_compact(const float* __restrict__ scores, uint32_t* __restrict__ meta,
                          unsigned long long* __restrict__ cand, int N) {
    int i = blockIdx.x * blockDim.x + threadIdx.x;
    if (i >= N) return;
    float s = scores[i];
    if (s < THR_CONF) return;
    uint32_t b = __float_as_uint(s);
    uint32_t bin = (b - 0x3F000000u) >> 3;
    if (bin >= NBINS) bin = NBINS - 1u;
    uint32_t T = meta[0], cA = meta[1];
    uint32_t pos;
    if (bin > T)       pos = atomicAdd(&meta[2], 1u);
    else if (bin == T) pos = cA + atomicAdd(&meta[3], 1u);
    else return;
    if (pos < CAND) {
        // key: score bits descending, index ascending (via complement)
        cand[pos] = ((unsigned long long)b << 32) | (unsigned long long)(0xFFFFFFFFu - (uint32_t)i);
    }
}

// ---------------- K4: one-block bitonic sort of 8192 keys, emit top-K ----------------
__global__ void __launch_bounds__(1024) k_sort(const unsigned long long* __restrict__ cand,
                                               uint2* __restrict__ top, float* __restrict__ out,
                                               int N, int A) {
    __shared__ unsigned long long keys[CAND];   // 64 KiB LDS
    int t = threadIdx.x;
    for (int i = t; i < CAND; i += 1024) keys[i] = cand[i];
    __syncthreads();
    for (int k = 2; k <= CAND; k <<= 1) {
        for (int j = k >> 1; j > 0; j >>= 1) {
            for (int i = t; i < CAND; i += 1024) {
                int l = i ^ j;
                if (l > i) {
                    unsigned long long a = keys[i], bb = keys[l];
                    bool up = (i & k) == 0;                 // descending run
                    bool sw = up ? (a < bb) : (a > bb);
                    if (sw) { keys[i] = bb; keys[l] = a; }
                }
            }
            __syncthreads();
        }
    }
    for (int i = t; i < KTOP; i += 1024) {
        unsigned long long key = keys[i];
        uint32_t bits = (uint32_t)(key >> 32);
        uint32_t idx  = 0xFFFFFFFFu - (uint32_t)(key & 0xFFFFFFFFull);
        if (idx >= (uint32_t)N) idx = 0u;                  // pad entries (never in practice)
        uint2 e; e.x = bits; e.y = idx;
        top[i] = e;
        uint32_t img = idx / (uint32_t)A;
        out[i]            = (float)img;                    // top_idxs - 1
        out[5 * KTOP + i] = 1.0f;                          // p_labels
        out[6 * KTOP + i] = __uint_as_float(bits);         // top_scores
    }
}

// ---------------- K5: gather-decode boxes for the 4096 survivors ----------------
__global__ void k_decode(const uint2* __restrict__ top, const float* __restrict__ p_loc,
                         const float* __restrict__ anchors, float* __restrict__ out,
                         float* __restrict__ nmsb, int A) {
    int t = blockIdx.x * blockDim.x + threadIdx.x;
    if (t >= KTOP) return;
    uint32_t f = top[t].y;
    uint32_t img = f / (uint32_t)A;
    uint32_t a = f - img * (uint32_t)A;
    float cx = anchors[4 * a], cy = anchors[4 * a + 1];
    float w  = anchors[4 * a + 2], h = anchors[4 * a + 3];
    float p0 = p_loc[4 * (size_t)f], p1 = p_loc[4 * (size_t)f + 1];
    float p2 = p_loc[4 * (size_t)f + 2], p3 = p_loc[4 * (size_t)f + 3];
    float ncx = cx + p0 * 0.1f * w, ncy = cy + p1 * 0.1f * h;
    float nw = w * expf(p2 * 0.2f), nh = h * expf(p3 * 0.2f);
    float l = ncx - nw * 0.5f, tt = ncy - nh * 0.5f;
    float r = ncx + nw * 0.5f, bb = ncy + nh * 0.5f;
    out[KTOP + 4 * t]     = l;  out[KTOP + 4 * t + 1] = tt;
    out[KTOP + 4 * t + 2] = r;  out[KTOP + 4 * t + 3] = bb;
    float off = (float)(img + 1u) * IMG_OFF;               // batched-NMS trick
    nmsb[4 * t]     = l + off;  nmsb[4 * t + 1] = tt + off;
    nmsb[4 * t + 2] = r + off;  nmsb[4 * t + 3] = bb + off;
}

// ---------------- K6: suppression bitmask; TDM stage + WMMA area-sums ----------------
// Grid: 2048 blocks x 256 threads (8 waves), 64 KiB *dynamic* LDS per block.
// Dynamic LDS (external addrspace(3) global) keeps the compiler from folding
// shb[] loads to undef even though the only writer is the TDM engine.
__global__ void __launch_bounds__(256) k_iou_mask(const float* __restrict__ nmsb,
                                                  unsigned long long* __restrict__ mask) {
    extern __shared__ float shb[];                         // 64 KiB, LDS offset 0
    int wv = (int)(threadIdx.x >> 5);
    int lane = (int)(threadIdx.x & 31);

    // Never-taken (but unprovable) store: guarantees a visible writer of shb.
    if (blockIdx.x == 0xFFFFFFFFu) shb[lane] = 0.0f;

    if (wv == 0) {
        // Tensor DMA descriptor: 1-D copy of 8192 x 8B from nmsb -> LDS offset 0
        uint64_t g = (uint64_t)(uintptr_t)nmsb;
        v4u g0;
        g0[0] = 1u;                                        // count=1, valid user D#
        g0[1] = 0u;                                        // lds_addr = 0
        g0[2] = (uint32_t)(g & 0xFFFFFFFFull);             // global_addr[31:0]
        g0[3] = ((uint32_t)(g >> 32) & 0x01FFFFFFu) | 0x80000000u;  // addr[56:32] | type=2
        const uint32_t td0 = (KTOP * 4u * 4u) / 8u;        // 8192 elements of 8 bytes
        v8i g1;
        g1[0] = (int)(3u << 16);                           // data_size = 8B
        g1[1] = (int)((td0 & 0xFFFFu) << 16);              // tensor_dim0[15:0]
        g1[2] = (int)((td0 >> 16) | (1u << 16));           // tensor_dim0[31:16] | tensor_dim1=1
        g1[3] = (int)((td0 & 0xFFFFu) << 16);              // tile_dim0 = 8192
        g1[4] = 1;                                         // tile_dim1 = 1
        g1[5] = (int)td0;                                  // tensor_dim0_stride
        g1[6] = 0; g1[7] = 0;
        v4i z; z[0] = 0; z[1] = 0; z[2] = 0; z[3] = 0;
        v8i z8; z8[0] = 0; z8[1] = 0; z8[2] = 0; z8[3] = 0;
        z8[4] = 0; z8[5] = 0; z8[6] = 0; z8[7] = 0;
        // 6-arg form (clang-23 / therock-10.0 headers): groups 0-3 + ext group + cpol
        __builtin_amdgcn_tensor_load_to_lds(g0, g1, z, z, z8, 0);
        __builtin_amdgcn_s_wait_tensorcnt(0);
    }
    __syncthreads();
    asm volatile("" ::: "memory");                         // order LDS reads after staging

    int gw = (int)blockIdx.x * 8 + wv;                     // 16384 waves total
    int rowTile = gw >> 6;                                 // 0..255
    int colWord = gw & 63;                                 // 0..63
    int i0 = rowTile * 16;
    int j0 = colWord * 64;
    int M = lane & 15;
    int half = lane >> 4;

    // A operand (16x4 f32): lanes 0-15 hold K=0,1 ; lanes 16-31 hold K=2,3 (zeros)
    const float* bi = &shb[(i0 + M) * 4];
    float areaRow = (bi[2] - bi[0]) * (bi[3] - bi[1]);
    v2f av;
    av.x = (half == 0) ? areaRow : 0.0f;                   // A[M,0]=area_i, A[M,2]=0
    av.y = (half == 0) ? 1.0f : 0.0f;                      // A[M,1]=1,      A[M,3]=0

    unsigned long long accLo[8] = {0, 0, 0, 0, 0, 0, 0, 0};
    unsigned long long accHi[8] = {0, 0, 0, 0, 0, 0, 0, 0};

    for (int c = 0; c < 4; ++c) {
        int jb = j0 + c * 16;
        int j = jb + M;                                    // this lane's column
        const float* pj = &shb[j * 4];
        float jl = pj[0], jt = pj[1], jr = pj[2], jbt = pj[3];
        float areaCol = (jr - jl) * (jbt - jt);
        // B operand (4x16 f32): lanes 0-15 hold K=0,1 ; lanes 16-31 hold K=2,3 (zeros)
        v2f bv;
        bv.x = (half == 0) ? 1.0f : 0.0f;                  // B[0][N]=1
        bv.y = (half == 0) ? areaCol : 0.0f;               // B[1][N]=area_j
        v8f cz = {0.f, 0.f, 0.f, 0.f, 0.f, 0.f, 0.f, 0.f};
        // D[M][N] = area_i + area_j on the matrix unit, co-executing with VALU below
        v8f d = __builtin_amdgcn_wmma_f32_16x16x4_f32(false, av, false, bv,
                                                      (short)0, cz, false, false);
#pragma unroll
        for (int r = 0; r < 8; ++r) {
            int i = i0 + r + 8 * half;                     // D vgpr r -> row M = r + 8*half
            const float* pi = &shb[i * 4];
            float iw = fminf(pi[2], jr) - fmaxf(pi[0], jl);
            float ih = fminf(pi[3], jbt) - fmaxf(pi[1], jt);
            iw = fmaxf(iw, 0.0f); ih = fmaxf(ih, 0.0f);
            float inter = iw * ih;
            float uni = d[r] - inter + 1e-9f;              // > 0 always
            // iou > thr  <=>  inter > thr * union   (division-free)
            bool pred = (inter > THR_NMS * uni) && (j > i);
            unsigned int bal = __builtin_amdgcn_ballot_w32(pred);
            accLo[r] |= (unsigned long long)(bal & 0xFFFFu) << (16 * c);  // row i0+r
            accHi[r] |= (unsigned long long)(bal >> 16) << (16 * c);      // row i0+r+8
        }
    }
    if (lane == 0) {
#pragma unroll
        for (int r = 0; r < 8; ++r) {
            mask[(unsigned long long)(i0 + r) * 64ull + (unsigned long long)colWord] = accLo[r];
            mask[(unsigned long long)(i0 + r + 8) * 64ull + (unsigned long long)colWord] = accHi[r];
        }
    }
}

// ---------------- K7: serial greedy reduce over the bitmask ----------------
__global__ void k_nms_reduce(const unsigned long long* __restrict__ mask,
                             const uint2* __restrict__ top, float* __restrict__ out) {
    __shared__ unsigned long long removed[64];
    int t = threadIdx.x;                                   // 64 threads
    removed[t] = 0ull;
    __syncthreads();
    for (int i = 0; i < KTOP; ++i) {
        bool sup = (removed[i >> 6] >> (i & 63)) & 1ull;
        __syncthreads();
        if (!sup) removed[t] |= mask[(unsigned long long)i * 64ull + (unsigned long long)t];
        __syncthreads();
    }
    for (int i = t; i < KTOP; i += 64) {
        bool sup = (removed[i >> 6] >> (i & 63)) & 1ull;
        bool pos = top[i].x != 0u;                         // score > 0
        out[7 * KTOP + i] = (!sup && pos) ? 1.0f : 0.0f;
    }
}

// ---------------- host launcher ----------------
extern "C" void kernel_launch(void* const* d_in, const int* in_sizes, int n_in,
                              void* d_out, int out_size, void* d_ws, size_t ws_size,
                              hipStream_t stream) {
    const float* p_loc   = (const float*)d_in[0];
    const float* p_conf  = (const float*)d_in[1];
    const float* anchors = (const float*)d_in[2];
    float* out = (float*)d_out;

    const int A = in_sizes[2] / 4;       // 131072
    const int N = in_sizes[1] / 2;       // B*A = 4,194,304

    uint8_t* w = (uint8_t*)d_ws;
    float*              scores = (float*)(w + OFF_SCORES);
    uint32_t*           hist   = (uint32_t*)(w + OFF_HIST);
    uint32_t*           meta   = (uint32_t*)(w + OFF_META);
    unsigned long long* cand   = (unsigned long long*)(w + OFF_CAND);
    uint2*              top    = (uint2*)(w + OFF_TOP);
    float*              nmsb   = (float*)(w + OFF_NMSB);
    unsigned long long* mask   = (unsigned long long*)(w + OFF_MASK);

    // hist + meta + cand are contiguous: zero in one pass
    const int zwords = (int)NBINS + 16 + CAND * 2;
    k_init<<<(zwords + 255) / 256, 256, 0, stream>>>(hist, zwords);

    k_score_hist<<<(N + 255) / 256, 256, 0, stream>>>(p_conf, scores, hist, N);
    k_thresh<<<1, 1024, 0, stream>>>(hist, meta);
    k_compact<<<(N + 255) / 256, 256, 0, stream>>>(scores, meta, cand, N);
    k_sort<<<1, 1024, 0, stream>>>(cand, top, out, N, A);
    k_decode<<<(KTOP + 255) / 256, 256, 0, stream>>>(top, p_loc, anchors, out, nmsb, A);
    k_iou_mask<<<2048, 256, 65536, stream>>>(nmsb, mask);   // 64 KiB dynamic LDS
    k_nms_reduce<<<1, 64, 0, stream>>>(mask, top, out);
}